// NonMaxSuppression_36979668418762
// MI455X (gfx1250) — compile-verified
//
#include <hip/hip_runtime.h>
#include <hip/hip_bf16.h>
#include <stdint.h>

// Problem constants (from reference setup_inputs / module constants).
constexpr int   Bn   = 16;     // batches
constexpr int   Nn   = 2048;   // boxes per batch
constexpr int   Cn   = 80;     // classes
constexpr int   Kn   = 100;    // max det per class
constexpr int   Mn   = 100;    // max det total
constexpr int   Tn   = 256;    // threads per block
constexpr int   IPT  = Nn / Tn;   // 8 items per thread
constexpr float CONF = 0.05f;
constexpr float IOUT = 0.5f;
constexpr float Wf   = 640.0f;
constexpr float Hf   = 640.0f;

__device__ __forceinline__ void argmax_pair(float& s, int& i, float os, int oi) {
  // max score, ties -> lowest index (matches jnp.argmax / lax.top_k ordering)
  if (os > s || (os == s && oi < i)) { s = os; i = oi; }
}

__device__ __forceinline__ void wave_argmax(float& s, int& i) {
#pragma unroll
  for (int off = 16; off > 0; off >>= 1) {
    float os = __shfl_xor(s, off, 32);
    int   oi = __shfl_xor(i, off, 32);
    argmax_pair(s, i, os, oi);
  }
}

// ---------------------------------------------------------------------------
// Stage 1: one block per (batch, class). Greedy NMS over 2048 boxes.
// Predictions for the batch are staged into LDS via the CDNA5 async
// global->LDS DMA path (ASYNCcnt), converted to rel_yxyx in place; per-lane
// scores live in registers; argmax = wave32 shuffles + tiny LDS combine.
// ---------------------------------------------------------------------------
__global__ __launch_bounds__(Tn) void nms_per_class_kernel(
    const float* __restrict__ pred,   // [B,N,6]
    float* __restrict__ sc_f,         // [B,C,K] workspace scores
    int* __restrict__ idx_f) {        // [B,C,K] workspace indices
  const int c = blockIdx.x;
  const int b = blockIdx.y;
  const int t = threadIdx.x;

  __shared__ float predBuf[Nn * 6];          // 48 KB
  __shared__ float redS[Tn / 32];
  __shared__ int   redI[Tn / 32];
  __shared__ float wS;
  __shared__ int   wI;

  // Zero this block's output slice (harness does not re-poison between
  // replays; the tail past early-exit must be exact zeros like the reference).
  const int slice = (b * Cn + c) * Kn;
  if (t < Kn) { sc_f[slice + t] = 0.0f; idx_f[slice + t] = 0; }

  // ---- async DMA: predictions[b] (48 KB) global -> LDS (ASYNCcnt path) ----
  {
    const float* gbase = pred + (size_t)b * Nn * 6;
    uint32_t ldsBase = (uint32_t)(uintptr_t)(&predBuf[0]);
    const int CHUNKS = (Nn * 6 * 4) / 16;   // 3072 x b128
    for (int ch = t; ch < CHUNKS; ch += Tn) {
      uint64_t ga = (uint64_t)(uintptr_t)gbase + (uint64_t)ch * 16u;
      uint32_t la = ldsBase + (uint32_t)ch * 16u;
      asm volatile("global_load_async_to_lds_b128 %0, %1, off"
                   :: "v"(la), "v"(ga) : "memory");
    }
    asm volatile("s_wait_asynccnt 0x0" ::: "memory");
  }
  __syncthreads();

  // Convert center_xyWH -> rel_yxyx in place; thresholded per-class scores in
  // registers. Stride-6 float LDS layout is bank-conflict-free for 32 lanes.
  float st[IPT], sl[IPT], sb[IPT], sr[IPT], sc[IPT];
#pragma unroll
  for (int q = 0; q < IPT; ++q) {
    const int i = t + q * Tn;
    float cx = predBuf[i * 6 + 0], cy = predBuf[i * 6 + 1];
    float w_ = predBuf[i * 6 + 2], h_ = predBuf[i * 6 + 3];
    float cl = predBuf[i * 6 + 4], sv = predBuf[i * 6 + 5];
    float top   = (cy - h_ * 0.5f) / Hf;
    float left  = (cx - w_ * 0.5f) / Wf;
    float bot   = (cy + h_ * 0.5f) / Hf;
    float right = (cx + w_ * 0.5f) / Wf;
    st[q] = top; sl[q] = left; sb[q] = bot; sr[q] = right;
    float s = ((int)cl == c) ? sv : 0.0f;     // score * onehot
    sc[q] = (s > CONF) ? s : -1.0f;           // conf threshold -> -1
    predBuf[i * 6 + 0] = top;  predBuf[i * 6 + 1] = left;
    predBuf[i * 6 + 2] = bot;  predBuf[i * 6 + 3] = right;
  }
  __syncthreads();

  for (int k = 0; k < Kn; ++k) {
    // block argmax of scores (regs -> wave32 shuffle -> 8-entry LDS combine)
    float bs = -INFINITY; int bi = 0;
#pragma unroll
    for (int q = 0; q < IPT; ++q) argmax_pair(bs, bi, sc[q], t + q * Tn);
    wave_argmax(bs, bi);
    if ((t & 31) == 0) { redS[t >> 5] = bs; redI[t >> 5] = bi; }
    __syncthreads();
    if (t == 0) {
      float s0 = redS[0]; int i0 = redI[0];
#pragma unroll
      for (int w = 1; w < Tn / 32; ++w) argmax_pair(s0, i0, redS[w], redI[w]);
      wS = s0; wI = i0;
      if (s0 > 0.0f) { sc_f[slice + k] = s0; idx_f[slice + k] = i0; }
    }
    __syncthreads();
    const float wsc = wS; const int wid = wI;
    if (wsc <= 0.0f) break;   // exact: reference tail records idx=0/score=0

    // suppress: IoU of winner vs this lane's 8 boxes (all register math)
    const float wt = predBuf[wid * 6 + 0], wl = predBuf[wid * 6 + 1];
    const float wb = predBuf[wid * 6 + 2], wr = predBuf[wid * 6 + 3];
    const float wa = (wb - wt) * (wr - wl);
#pragma unroll
    for (int q = 0; q < IPT; ++q) {
      const int i = t + q * Tn;
      float y1 = fmaxf(wt, st[q]), x1 = fmaxf(wl, sl[q]);
      float y2 = fminf(wb, sb[q]), x2 = fminf(wr, sr[q]);
      float inter = fmaxf(y2 - y1, 0.0f) * fmaxf(x2 - x1, 0.0f);
      float ar  = (sb[q] - st[q]) * (sr[q] - sl[q]);
      float un  = wa + ar - inter;
      float iou = (un > 0.0f) ? (inter / un) : 0.0f;
      if (iou > IOUT || i == wid) sc[q] = -1.0f;
    }
  }
}

// ---------------------------------------------------------------------------
// Stage 2: one block per batch. Top-100 of the 8000 (class,slot) scores via
// 100 LDS-resident argmax sweeps; reconstruct output rows exactly along the
// reference's float path (center->rel_yxyx->center).
// ---------------------------------------------------------------------------
__global__ __launch_bounds__(Tn) void topk_finalize_kernel(
    const float* __restrict__ pred,   // [B,N,6]
    const float* __restrict__ sc_f,   // [B,C,K]
    const int* __restrict__ idx_f,    // [B,C,K]
    float* __restrict__ out) {        // [B,M,6] then valid[B] (as float)
  const int b = blockIdx.x;
  const int t = threadIdx.x;
  constexpr int F = Cn * Kn;          // 8000

  __shared__ float s_sc[F];           // ~31.3 KB
  __shared__ float redS[Tn / 32];
  __shared__ int   redI[Tn / 32];

  for (int j = t; j < F; j += Tn) s_sc[j] = sc_f[(size_t)b * F + j];
  __syncthreads();

  int cnt = 0;   // only thread 0's copy matters
  for (int m = 0; m < Mn; ++m) {
    float bs = -INFINITY; int bi = 0;
    for (int j = t; j < F; j += Tn) argmax_pair(bs, bi, s_sc[j], j);
    wave_argmax(bs, bi);
    if ((t & 31) == 0) { redS[t >> 5] = bs; redI[t >> 5] = bi; }
    __syncthreads();
    if (t == 0) {
      float s0 = redS[0]; int i0 = redI[0];
#pragma unroll
      for (int w = 1; w < Tn / 32; ++w) argmax_pair(s0, i0, redS[w], redI[w]);
      float* row = out + ((size_t)b * Mn + m) * 6;
      if (s0 > 0.0f) {
        const int cls  = i0 / Kn;
        const int bidx = idx_f[(size_t)b * F + i0];
        const float* p = pred + ((size_t)b * Nn + bidx) * 6;
        float cx = p[0], cy = p[1], w_ = p[2], h_ = p[3];
        float top   = (cy - h_ * 0.5f) / Hf;
        float left  = (cx - w_ * 0.5f) / Wf;
        float bot   = (cy + h_ * 0.5f) / Hf;
        float right = (cx + w_ * 0.5f) / Wf;
        row[0] = (left + right) * 0.5f * Wf;
        row[1] = (top + bot) * 0.5f * Hf;
        row[2] = (right - left) * Wf;
        row[3] = (bot - top) * Hf;
        row[4] = (float)cls;
        row[5] = s0;
        cnt++;
        s_sc[i0] = -1.0f;   // exclude from subsequent sweeps
      } else {
        row[0] = row[1] = row[2] = row[3] = row[4] = row[5] = 0.0f;
      }
    }
    __syncthreads();
  }
  if (t == 0) out[(size_t)Bn * Mn * 6 + b] = (float)cnt;  // valid count
}

extern "C" void kernel_launch(void* const* d_in, const int* in_sizes, int n_in,
                              void* d_out, int out_size, void* d_ws, size_t ws_size,
                              hipStream_t stream) {
  (void)in_sizes; (void)n_in; (void)out_size; (void)ws_size;
  // d_in[0] = images [16,640,640,3] f32 (all zeros; only shape matters — unused)
  // d_in[1] = predictions [16,2048,6] f32
  const float* pred = (const float*)d_in[1];
  float* out = (float*)d_out;

  // Workspace: scores [B*C*K] f32, then indices [B*C*K] i32 (1 MB total).
  float* sc_f = (float*)d_ws;
  int*   idx_f = (int*)((char*)d_ws + (size_t)Bn * Cn * Kn * sizeof(float));

  dim3 g1(Cn, Bn);                 // 80 x 16 = 1280 independent NMS tasks
  nms_per_class_kernel<<<g1, Tn, 0, stream>>>(pred, sc_f, idx_f);
  topk_finalize_kernel<<<Bn, Tn, 0, stream>>>(pred, sc_f, idx_f, out);
}